// NormalizedCrossCorrelationLoss_78417512890776
// MI455X (gfx1250) — compile-verified
//
#include <hip/hip_runtime.h>

// NCC loss, window 9, shape (4,1,160,160,160) fp32.
// Fused: one workgroup per 16x16 xy tile, sliding along z.
// x/y box sums on the matrix engine (V_WMMA_F32_16X16X4_F32, banded 0/1
// weights), z box sum via 9-slice LDS ring buffer. Halo slices staged into
// LDS by the Tensor Data Mover (tensor_load_to_lds) for interior tiles.

#define D        160
#define TILE     16
#define HALO     4
#define HW       24          // TILE + 2*HALO
#define NQ       5           // p, t, p^2, t^2, p*t
#define NTHREADS 160         // 5 waves (wave32): one wave per quantity
#define WINSZ    729.0f
#define NVOX     16384000.0  // 4 * 160^3

typedef __attribute__((ext_vector_type(2))) float v2f;
typedef __attribute__((ext_vector_type(8))) float v8f;
typedef __attribute__((ext_vector_type(4))) unsigned int v4u;
typedef __attribute__((ext_vector_type(8))) int v8i;
typedef __attribute__((ext_vector_type(4))) int v4i;

#if __has_builtin(__builtin_amdgcn_tensor_load_to_lds) && \
    __has_builtin(__builtin_amdgcn_s_wait_tensorcnt)
#define USE_TDM 1
#else
#define USE_TDM 0
#endif

__global__ __launch_bounds__(NTHREADS)
void ncc_main(const float* __restrict__ pred, const float* __restrict__ targ,
              double* __restrict__ accum) {
  __shared__ float sP[HW][HW];                // pred halo slice (row = 96B, so
  __shared__ float sT[HW][HW];                //  even-col v2f loads stay aligned)
  __shared__ float sMid[NQ][HW][TILE + 1];    // after x box sum (24 y-rows)
  __shared__ float sRing[NQ][9][TILE * TILE]; // xy-box-summed slices, ring on z
  __shared__ float sRed[NTHREADS];

  const int tid  = threadIdx.x;
  const int wave = tid >> 5;       // quantity owned by this wave
  const int lane = tid & 31;
  const int hi   = lane >> 4;      // half-wave (ISA WMMA f32 layouts)
  const int n    = lane & 15;      // per-lane M (A/D rows) == N (B/D cols)

  const int x0 = blockIdx.x * TILE;
  const int y0 = blockIdx.y * TILE;
  const long basev = (long)blockIdx.z * D * D * D;

  // branchless per-wave quantity select: val = fa * fb
  //   q: 0->p*1  1->t*1  2->p*p  3->t*t  4->p*t
  const bool selAT = (wave == 1) || (wave == 3);  // fa = t (else p)
  const bool selB1 = (wave <= 1);                 // fb = 1
  const bool selBT = (wave >= 3);                 // fb = t (else p), if !selB1

#if USE_TDM
  const bool tdmOK = (x0 >= HALO) && (y0 >= HALO);
  // ---- D# group1 (constant per block): 2D tile 24x24 of 4-byte elems ----
  const unsigned td0 = (unsigned)(D + HALO - x0);  // remaining x extent ->
  const unsigned td1 = (unsigned)(D + HALO - y0);  //  right/bottom zero-fill
  v8i g1;
  g1[0] = (int)(2u << 16);                         // data_size=4B, mask=0
  g1[1] = (int)((td0 & 0xFFFFu) << 16);            // tensor_dim0[15:0]
  g1[2] = (int)((td0 >> 16) | ((td1 & 0xFFFFu) << 16));
  g1[3] = (int)((td1 >> 16) | ((unsigned)HW << 16));  // tile_dim0 = 24
  g1[4] = (int)((unsigned)HW);                     // tile_dim1 = 24, tile_dim2=0
  g1[5] = (int)((unsigned)D);                      // tensor_dim0_stride = 160
  g1[6] = 0;
  g1[7] = 0;
  const v4i gz4 = {0, 0, 0, 0};                    // groups 2/3 unused (2D)
  const v8i gz8 = {0, 0, 0, 0, 0, 0, 0, 0};        // extra group (unused)
  const unsigned ldsP = (unsigned)(uintptr_t)&sP[0][0];
  const unsigned ldsT = (unsigned)(uintptr_t)&sT[0][0];
#endif

  // zero ring buffer (slots for z_in = -4..-1 must read as zero padding)
  for (int i = tid; i < NQ * 9 * 256; i += NTHREADS)
    ((float*)sRing)[i] = 0.f;
  __syncthreads();

  float local = 0.f;

  for (int zi = 0; zi < D + HALO; ++zi) {      // zi = input z plane
    const int slot = zi % 9;
    if (zi < D) {
      // ---- stage halo slice of pred/target into LDS ----
      const long basez = basev + (long)zi * D * D;
#if USE_TDM
      if (tdmOK) {
        const long horg = basez + (long)(y0 - HALO) * D + (x0 - HALO);
        if (wave == 0) {
          const unsigned long long ga = (unsigned long long)(uintptr_t)(pred + horg);
          v4u g0;
          g0[0] = 1u;                            // count=1, no gather
          g0[1] = ldsP;
          g0[2] = (unsigned)(ga & 0xFFFFFFFFu);
          g0[3] = (unsigned)((ga >> 32) & 0x1FFFFFFu) | (2u << 30);  // type=2
          __builtin_amdgcn_tensor_load_to_lds(g0, g1, gz4, gz4, gz8, 0);
          __builtin_amdgcn_s_wait_tensorcnt(0);
        } else if (wave == 1) {
          const unsigned long long ga = (unsigned long long)(uintptr_t)(targ + horg);
          v4u g0;
          g0[0] = 1u;
          g0[1] = ldsT;
          g0[2] = (unsigned)(ga & 0xFFFFFFFFu);
          g0[3] = (unsigned)((ga >> 32) & 0x1FFFFFFu) | (2u << 30);
          __builtin_amdgcn_tensor_load_to_lds(g0, g1, gz4, gz4, gz8, 0);
          __builtin_amdgcn_s_wait_tensorcnt(0);
        }
      } else
#endif
      {
        for (int e = tid; e < HW * HW; e += NTHREADS) {
          const int hy = e / HW, hx = e - hy * HW;
          const int gy = y0 - HALO + hy, gx = x0 - HALO + hx;
          float pv = 0.f, tv = 0.f;
          if (gy >= 0 && gy < D && gx >= 0 && gx < D) {
            const long idx = basez + (long)gy * D + gx;
            pv = pred[idx];
            tv = targ[idx];
            if (e == tid && zi + 8 < D) {        // global_prefetch_b8
              __builtin_prefetch(&pred[idx + 8 * D * D], 0, 1);
              __builtin_prefetch(&targ[idx + 8 * D * D], 0, 1);
            }
          }
          sP[hy][hx] = pv;
          sT[hy][hx] = tv;
        }
      }
      __syncthreads();

      const int q = wave;
      // ---- x box sum: Mid(24x16) = In(24x24) x Wx(24x16), banded 0/1 ----
      for (int rb = 0; rb < 2; ++rb) {
        const int ry = rb * 8;
        v8f acc = {};
#pragma unroll
        for (int c = 0; c < 6; ++c) {            // K = 24 -> 6 x (K=4)
          const int k0 = c * 4 + 2 * hi;         // even -> 8B-aligned LDS pair
          const v2f pp = *(const v2f*)&sP[ry + n][k0];
          const v2f tt = *(const v2f*)&sT[ry + n][k0];
          v2f a, bb;
          {
            const float fax = selAT ? tt.x : pp.x;
            const float fbx = selB1 ? 1.f : (selBT ? tt.x : pp.x);
            const float fay = selAT ? tt.y : pp.y;
            const float fby = selB1 ? 1.f : (selBT ? tt.y : pp.y);
            a.x = fax * fbx;
            a.y = fay * fby;
          }
          bb.x = (n <= k0 && k0 <= n + 8) ? 1.f : 0.f;      // Wx[k][n]
          bb.y = (n <= k0 + 1 && k0 + 1 <= n + 8) ? 1.f : 0.f;
          acc = __builtin_amdgcn_wmma_f32_16x16x4_f32(
              false, a, false, bb, (short)0, acc, false, false);
        }
#pragma unroll
        for (int r = 0; r < 8; ++r) {
          const int row = ry + r + 8 * hi;       // D layout: M = r + 8*hi
          if (rb == 0 || row >= 16) sMid[q][row][n] = acc[r];
        }
      }
      // ---- y box sum: Out(16x16) = Wy(16x24) x Mid(24x16) ----
      {
        v8f acc = {};
#pragma unroll
        for (int c = 0; c < 6; ++c) {
          const int k0 = c * 4 + 2 * hi;
          v2f a, bb;
          a.x  = (n <= k0 && k0 <= n + 8) ? 1.f : 0.f;      // Wy[m][k]
          a.y  = (n <= k0 + 1 && k0 + 1 <= n + 8) ? 1.f : 0.f;
          bb.x = sMid[q][k0][n];
          bb.y = sMid[q][k0 + 1][n];
          acc = __builtin_amdgcn_wmma_f32_16x16x4_f32(
              false, a, false, bb, (short)0, acc, false, false);
        }
#pragma unroll
        for (int r = 0; r < 8; ++r) {
          const int y = r + 8 * hi;
          sRing[q][slot][y * TILE + n] = acc[r];
        }
      }
    } else {
      // slice entirely past the volume: zero this ring slot (SAME padding)
      for (int i = tid; i < NQ * 256; i += NTHREADS)
        sRing[i >> 8][slot][i & 255] = 0.f;
    }

    if (zi >= HALO) {
      __syncthreads();
      // ring now holds exactly z_out-4 .. z_out+4 (out-of-range = zeros)
      for (int v = tid; v < 256; v += NTHREADS) {
        float s0 = 0.f, s1 = 0.f, s2 = 0.f, s3 = 0.f, s4 = 0.f;
#pragma unroll
        for (int sl = 0; sl < 9; ++sl) {
          s0 += sRing[0][sl][v];
          s1 += sRing[1][sl][v];
          s2 += sRing[2][sl][v];
          s3 += sRing[3][sl][v];
          s4 += sRing[4][sl][v];
        }
        const float pm = s0 * (1.f / WINSZ);
        const float tm = s1 * (1.f / WINSZ);
        const float cross = s4 - tm * s0 - pm * s1 + pm * tm * WINSZ;
        const float pvar  = s2 - 2.f * pm * s0 + pm * pm * WINSZ;
        const float tvar  = s3 - 2.f * tm * s1 + tm * tm * WINSZ;
        local += cross * cross / (pvar * tvar + 1e-8f);
      }
    }
    __syncthreads();  // protect sP/sT and the oldest ring slot for next zi
  }

  // block reduction -> one double atomic per workgroup
  sRed[tid] = local;
  __syncthreads();
  if (tid == 0) {
    double bs = 0.0;
    for (int i = 0; i < NTHREADS; ++i) bs += sRed[i];
    atomicAdd(accum, bs);
  }
}

__global__ void ncc_init(double* accum) {
  if (threadIdx.x == 0) *accum = 0.0;
}

__global__ void ncc_final(const double* __restrict__ accum,
                          float* __restrict__ out) {
  if (threadIdx.x == 0) out[0] = 1.0f - (float)(*accum / NVOX);
}

extern "C" void kernel_launch(void* const* d_in, const int* in_sizes, int n_in,
                              void* d_out, int out_size, void* d_ws, size_t ws_size,
                              hipStream_t stream) {
  const float* pred = (const float*)d_in[0];
  const float* targ = (const float*)d_in[1];
  double* accum = (double*)d_ws;
  float* out = (float*)d_out;

  hipLaunchKernelGGL(ncc_init, dim3(1), dim3(32), 0, stream, accum);
  dim3 grid(D / TILE, D / TILE, 4);
  hipLaunchKernelGGL(ncc_main, grid, dim3(NTHREADS), 0, stream, pred, targ, accum);
  hipLaunchKernelGGL(ncc_final, dim3(1), dim3(32), 0, stream, accum, out);
}